// BayesianTorchModel_37022618092107
// MI455X (gfx1250) — compile-verified
//
#include <hip/hip_runtime.h>

// Bayesian layered-network sweep for MI455X (gfx1250).
//
// Serial 63-layer chain, ~86 MB total traffic (fits in 192 MB L2) and ~0.2 G
// VALU ops -> latency/sync bound, not compute bound. No matmul structure, so
// no WMMA; instead we use the CDNA5 Tensor Data Mover to broadcast each
// freshly-computed 16384-float marginal layer (64 KB) from L2 into every
// block's LDS, so the per-node parent gathers are LDS reads.

#define L_HIDDEN 63
#define M_NODES  16384
#define KPAR     4
#define CCFG     16
#define NBLK     32
#define NTHR     512   // NBLK*NTHR == M_NODES: one node per thread per layer

typedef __attribute__((ext_vector_type(4))) unsigned int v4u;
typedef __attribute__((ext_vector_type(8))) int          v8i;
typedef __attribute__((ext_vector_type(4))) int          v4i;

__device__ __forceinline__ float fast_sigmoid(float x) {
  // sigmoid(x) = 1 / (1 + exp2(-x*log2(e)))  -> v_exp_f32 + v_rcp_f32
  float e = __builtin_amdgcn_exp2f(x * -1.44269504088896340736f);
  return __builtin_amdgcn_rcpf(1.0f + e);
}

// TDM: DMA one full marginal layer (16384 f32 = 64 KB) global -> LDS.
// 1-D tensor descriptor per CDNA5 ISA (6-arg builtin).
__device__ __forceinline__ void tdm_load_layer_to_lds(const float* gsrc, float* lds_dst) {
  unsigned long long ga = (unsigned long long)(const void*)gsrc;
  unsigned lds_off = (unsigned)(unsigned long long)(void*)lds_dst; // generic LDS addr low 32b = offset

  v4u g0;
  g0.x = 1u;                                   // count=1 (valid), is_restore=0, gather off
  g0.y = lds_off;                              // lds_addr [63:32]
  g0.z = (unsigned)ga;                         // global_addr [95:64]
  g0.w = (unsigned)((ga >> 32) & 0x01FFFFFFull)// global_addr [120:96] (bits 56:32)
       | (2u << 30);                           // type=2 ("image")

  v8i g1;
  g1[0] = (int)(2u << 16);                     // data_size=2 -> 4 bytes; workgroup_mask=0
  g1[1] = (int)((unsigned)(M_NODES & 0xFFFF) << 16);           // tensor_dim0[15:0] @ bits 63:48
  g1[2] = (int)(((unsigned)M_NODES >> 16) | (1u << 16));       // tensor_dim0[31:16]; tensor_dim1=1
  g1[3] = (int)((unsigned)(M_NODES & 0xFFFF) << 16);           // tensor_dim1 hi=0; tile_dim0=16384
  g1[4] = (int)(1u | (1u << 16));              // tile_dim1=1, tile_dim2=1 (single row; safe
                                               // under both "0=unused" and loop-count readings)
  g1[5] = M_NODES;                             // tensor_dim0_stride low32 (data_size units)
  g1[6] = 0;                                   // stride hi16 / tensor_dim1_stride low16
  g1[7] = 0;

  v4i z4 = {0, 0, 0, 0};                       // groups 2/3 unused (<=2D tensor)
  v8i z8 = {0, 0, 0, 0, 0, 0, 0, 0};           // extra group (clang-23 6-arg form)
  __builtin_amdgcn_tensor_load_to_lds(g0, g1, z4, z4, z8, 0);
}

__global__ void reset_barriers_kernel(unsigned* cnt) {
  if (threadIdx.x < 64) cnt[threadIdx.x] = 0u;
}

__global__ void __launch_bounds__(NTHR, 1)
bayes_layers_kernel(const float* __restrict__ root_logits,
                    const float* __restrict__ layer_logits,
                    const int*   __restrict__ parent_idx,
                    float*       __restrict__ out,
                    unsigned*    __restrict__ barrier_cnt) {
  __shared__ float smem[M_NODES];              // full previous-layer marginals (64 KB)

  const int m = (int)(blockIdx.x * NTHR + threadIdx.x);   // 0..16383

  // Layer 0 (roots): marg = sigmoid(root_logits)
  out[m] = fast_sigmoid(root_logits[m]);

  // Warm L2/L0 for layer 0 inputs while the first grid barrier settles.
  __builtin_prefetch(layer_logits + (size_t)m * CCFG, 0, 0);
  __builtin_prefetch(parent_idx + (size_t)m * KPAR, 0, 0);

  for (int l = 0; l < L_HIDDEN; ++l) {
    // ---- release stores of layer-l marginals, grid barrier (fresh counter per layer) ----
    __threadfence();
    __syncthreads();
    if (threadIdx.x == 0) {
      atomicAdd(&barrier_cnt[l], 1u);
      volatile unsigned* c = &barrier_cnt[l];
      while (*c < (unsigned)NBLK) { __builtin_amdgcn_s_sleep(1); }
    }
    __syncthreads();

    // ---- TDM refresh: broadcast layer-l marginals from L2 into this block's LDS ----
    if (threadIdx.x < 32u) {                   // wave 0 only issues the DMA
      tdm_load_layer_to_lds(out + (size_t)l * M_NODES, smem);
      __builtin_amdgcn_s_wait_tensorcnt(0);    // s_wait_tensorcnt 0
    }
    __syncthreads();

    // ---- gather 4 parent marginals from LDS ----
    const int4 pidx = *(const int4*)(parent_idx + ((size_t)l * M_NODES + m) * KPAR);
    const float p0 = smem[pidx.x], p1 = smem[pidx.y];
    const float p2 = smem[pidx.z], p3 = smem[pidx.w];
    const float q0 = 1.0f - p0, q1 = 1.0f - p1, q2 = 1.0f - p2, q3 = 1.0f - p3;

    // 2x2 butterfly pair products; config bit order is MSB-first (parent 0 = MSB)
    const float t01_0 = q0 * q1, t01_1 = q0 * p1, t01_2 = p0 * q1, t01_3 = p0 * p1;
    const float t23_0 = q2 * q3, t23_1 = q2 * p3, t23_2 = p2 * q3, t23_3 = p2 * p3;

    // ---- 16 CPT logits: 64 contiguous bytes per node ----
    const float4* lg = (const float4*)(layer_logits + ((size_t)l * M_NODES + m) * CCFG);
    const float4 v0 = lg[0], v1 = lg[1], v2 = lg[2], v3 = lg[3];

    float accA = 0.0f, accB = 0.0f;
    accA = fmaf(fast_sigmoid(v0.x), t01_0 * t23_0, accA);   // c=0
    accB = fmaf(fast_sigmoid(v0.y), t01_0 * t23_1, accB);   // c=1
    accA = fmaf(fast_sigmoid(v0.z), t01_0 * t23_2, accA);   // c=2
    accB = fmaf(fast_sigmoid(v0.w), t01_0 * t23_3, accB);   // c=3
    accA = fmaf(fast_sigmoid(v1.x), t01_1 * t23_0, accA);   // c=4
    accB = fmaf(fast_sigmoid(v1.y), t01_1 * t23_1, accB);   // c=5
    accA = fmaf(fast_sigmoid(v1.z), t01_1 * t23_2, accA);   // c=6
    accB = fmaf(fast_sigmoid(v1.w), t01_1 * t23_3, accB);   // c=7
    accA = fmaf(fast_sigmoid(v2.x), t01_2 * t23_0, accA);   // c=8
    accB = fmaf(fast_sigmoid(v2.y), t01_2 * t23_1, accB);   // c=9
    accA = fmaf(fast_sigmoid(v2.z), t01_2 * t23_2, accA);   // c=10
    accB = fmaf(fast_sigmoid(v2.w), t01_2 * t23_3, accB);   // c=11
    accA = fmaf(fast_sigmoid(v3.x), t01_3 * t23_0, accA);   // c=12
    accB = fmaf(fast_sigmoid(v3.y), t01_3 * t23_1, accB);   // c=13
    accA = fmaf(fast_sigmoid(v3.z), t01_3 * t23_2, accA);   // c=14
    accB = fmaf(fast_sigmoid(v3.w), t01_3 * t23_3, accB);   // c=15

    out[(size_t)(l + 1) * M_NODES + m] = accA + accB;

    // Prefetch NEXT layer's logits + indices now, so the L2 walk overlaps the
    // upcoming grid-barrier spin and TDM drain (the latency we're bound by).
    if (l + 1 < L_HIDDEN) {
      __builtin_prefetch(layer_logits + ((size_t)(l + 1) * M_NODES + m) * CCFG, 0, 0);
      __builtin_prefetch(parent_idx + ((size_t)(l + 1) * M_NODES + m) * KPAR, 0, 0);
    }
  }
}

extern "C" void kernel_launch(void* const* d_in, const int* in_sizes, int n_in,
                              void* d_out, int out_size, void* d_ws, size_t ws_size,
                              hipStream_t stream) {
  (void)in_sizes; (void)n_in; (void)out_size; (void)ws_size;
  const float* root_logits  = (const float*)d_in[0];   // [16384]
  const float* layer_logits = (const float*)d_in[1];   // [63,16384,16]
  const int*   parent_idx   = (const int*)  d_in[2];   // [63,16384,4]
  float*       out          = (float*)d_out;           // [64*16384]
  unsigned*    barrier_cnt  = (unsigned*)d_ws;         // 64 counters (256 B)

  // Zero grid-barrier counters every call -> deterministic under graph replay.
  reset_barriers_kernel<<<1, 64, 0, stream>>>(barrier_cnt);
  bayes_layers_kernel<<<NBLK, NTHR, 0, stream>>>(root_logits, layer_logits,
                                                 parent_idx, out, barrier_cnt);
}